// MultiHeadAttention_42820823941543
// MI455X (gfx1250) — compile-verified
//
#include <hip/hip_runtime.h>
#include <math.h>

// ---------------------------------------------------------------------------
// MI455X (gfx1250) fused MHA: fp32 WMMA (V_WMMA_F32_16X16X4_F32) everywhere,
// async global->LDS copies (ASYNCcnt) with LDS double buffering in the GEMMs.
// ---------------------------------------------------------------------------

typedef __attribute__((ext_vector_type(2))) float v2f;
typedef __attribute__((ext_vector_type(8))) float v8f;

#define D_MODEL 768
#define SEQ     2048
#define BATCH   4
#define HEADS   12
#define HDIM    64
#define ROWS    (BATCH * SEQ)   // 8192
#define GEMM_BN 128

__device__ __forceinline__ v8f wmma_f32(v2f a, v2f b, v8f c) {
  // (neg_a, A, neg_b, B, c_mod, C, reuse_a, reuse_b)
  return __builtin_amdgcn_wmma_f32_16x16x4_f32(
      false, a, false, b, (short)0, c, false, false);
}

__device__ __forceinline__ void async_copy_b128(const float* gsrc, void* ldst) {
  // GLOBAL_LOAD_ASYNC_TO_LDS_B128: per-lane 16B global -> LDS, ASYNCcnt.
  // Low 32 bits of a generic LDS pointer are the raw LDS offset (ISA 10.2).
  unsigned dst = (unsigned)(size_t)ldst;
  unsigned long long src = (unsigned long long)(size_t)gsrc;
  asm volatile("global_load_async_to_lds_b128 %0, %1, off"
               :: "v"(dst), "v"(src) : "memory");
}

__device__ __forceinline__ void wait_async_lds() {
  asm volatile("s_wait_asynccnt 0" ::: "memory");
}

// ---------------------------------------------------------------------------
// GEMM: C[M x 768] = A[M x 768] * B[768 x 768], fp32 WMMA.
// Block tile 64(M) x 128(N), 256 threads = 8 waves; each wave owns 32x32
// (4 accumulators: 2 A-frags x 2 B-frags -> 1 DS frag load per WMMA).
// K staged 16 at a time via async global->LDS copies, double-buffered so the
// next tile's fills overlap this tile's WMMAs.
// A kept row-major with stride-20 pad (conflict-free frag reads), B row-major.
// qkv_mode=1 scatters output into [b, h, n, d] layout for attention.
// ---------------------------------------------------------------------------
__global__ __launch_bounds__(256)
void gemm768_wmma(const float* __restrict__ A, const float* __restrict__ B,
                  float* __restrict__ C, int qkv_mode) {
  __shared__ float a_lds[2][64][20];    // [buf][m][k], pad 20 -> conflict-free
  __shared__ float b_lds[2][16][128];   // [buf][k][n]

  const int t    = threadIdx.x;
  const int lane = t & 31;
  const int wid  = t >> 5;
  const int half = lane >> 4;      // 0: lanes 0-15, 1: lanes 16-31
  const int Lm   = lane & 15;
  const int m_sub = (wid & 1) * 32;
  const int n_sub = (wid >> 1) * 32;
  const int blockN = blockIdx.x * GEMM_BN;
  const int blockM = blockIdx.y * 64;

  v8f c00 = {0,0,0,0,0,0,0,0};
  v8f c01 = {0,0,0,0,0,0,0,0};
  v8f c10 = {0,0,0,0,0,0,0,0};
  v8f c11 = {0,0,0,0,0,0,0,0};

  // staging assignment: A = 64 rows x 4 quads (1 per thread),
  //                     B = 16 rows x 32 quads (2 per thread)
  const int ar = t >> 2;           // 0..63
  const int aq = (t & 3) * 4;      // 0,4,8,12

  auto stage = [&](int k0, int buf) {
    async_copy_b128(&A[(size_t)(blockM + ar) * D_MODEL + k0 + aq],
                    &a_lds[buf][ar][aq]);
#pragma unroll
    for (int i = 0; i < 2; ++i) {
      const int idx = t + i * 256;
      const int kr  = idx >> 5;          // 0..15
      const int q   = (idx & 31) * 4;    // 0..124
      async_copy_b128(&B[(size_t)(k0 + kr) * D_MODEL + blockN + q],
                      &b_lds[buf][kr][q]);
    }
  };

  stage(0, 0);
  wait_async_lds();
  __syncthreads();

  for (int k0 = 0; k0 < D_MODEL; k0 += 16) {
    const int buf = (k0 >> 4) & 1;
    if (k0 + 16 < D_MODEL) stage(k0 + 16, buf ^ 1);   // overlap fill w/ WMMA

#pragma unroll
    for (int kk = 0; kk < 16; kk += 4) {
      const int krow = kk + 2 * half;   // lanes 0-15: K=kk,kk+1; 16-31: +2,+3
      v2f a0, a1, b0, b1;
      a0.x = a_lds[buf][m_sub + Lm][krow];
      a0.y = a_lds[buf][m_sub + Lm][krow + 1];
      a1.x = a_lds[buf][m_sub + 16 + Lm][krow];
      a1.y = a_lds[buf][m_sub + 16 + Lm][krow + 1];
      b0.x = b_lds[buf][krow][n_sub + Lm];
      b0.y = b_lds[buf][krow + 1][n_sub + Lm];
      b1.x = b_lds[buf][krow][n_sub + 16 + Lm];
      b1.y = b_lds[buf][krow + 1][n_sub + 16 + Lm];
      c00 = wmma_f32(a0, b0, c00);
      c01 = wmma_f32(a0, b1, c01);
      c10 = wmma_f32(a1, b0, c10);
      c11 = wmma_f32(a1, b1, c11);
    }
    wait_async_lds();
    __syncthreads();
  }

  // C/D layout: VGPR r, lanes 0-15 -> M=r, lanes 16-31 -> M=r+8; N = lane%16
  auto store_c = [&](int row, int col, float val) {
    if (qkv_mode) {
      const int bidx = row >> 11;          // row / 2048
      const int n    = row & 2047;
      const int hh   = col >> 6;           // col / 64
      const int d    = col & 63;
      C[((((size_t)bidx * HEADS + hh) * SEQ + n) << 6) + d] = val;
    } else {
      C[(size_t)row * D_MODEL + col] = val;
    }
  };
#pragma unroll
  for (int r = 0; r < 8; ++r) {
    const int row0 = blockM + m_sub + r + 8 * half;
    const int row1 = row0 + 16;
    const int col0 = blockN + n_sub + Lm;
    const int col1 = col0 + 16;
    store_c(row0, col0, c00[r]);
    store_c(row0, col1, c01[r]);
    store_c(row1, col0, c10[r]);
    store_c(row1, col1, c11[r]);
  }
}

// ---------------------------------------------------------------------------
// Flash attention, causal, head_dim = 64. One wave per 16-query tile per
// (b,h). Q frag register-resident in A-layout; K/V tiles staged in LDS
// (padded stride 68 -> conflict-free K^T fragment reads); online softmax with
// 16-lane xor reductions; P converted C-layout -> A-layout via LDS bounce.
// ---------------------------------------------------------------------------
__global__ __launch_bounds__(32)
void flash_attn_wmma(const float* __restrict__ Qm, const float* __restrict__ Km,
                     const float* __restrict__ Vm, float* __restrict__ ctx) {
  __shared__ float k_lds[16][68];
  __shared__ float v_lds[16][68];
  __shared__ float p_lds[16][17];

  const int lane = threadIdx.x;
  const int half = lane >> 4;
  const int Lm   = lane & 15;
  const int qt   = blockIdx.x;            // query tile (0..127)
  const int bh   = blockIdx.y;            // 0..47
  const int b    = bh / HEADS;
  const int h    = bh % HEADS;
  const size_t base = (size_t)bh * SEQ * HDIM;

  // Q tile (16 x 64) in WMMA A-layout: qf[i] covers K-dim [4i, 4i+3]
  v2f qf[16];
#pragma unroll
  for (int i = 0; i < 16; ++i) {
    const float* qp = &Qm[base + (size_t)(qt * 16 + Lm) * HDIM + i * 4 + 2 * half];
    qf[i].x = qp[0];
    qf[i].y = qp[1];
  }

  v8f acc[4];
#pragma unroll
  for (int i = 0; i < 4; ++i) acc[i] = (v8f){0,0,0,0,0,0,0,0};
  float mrow[8], lrow[8];
#pragma unroll
  for (int r = 0; r < 8; ++r) { mrow[r] = -INFINITY; lrow[r] = 0.0f; }

  for (int jt = 0; jt <= qt; ++jt) {
    // stage K, V tiles (16 x 64 each) cooperatively, b128 loads
#pragma unroll
    for (int i = 0; i < 8; ++i) {
      const int idx = i * 32 + lane;
      const int row = idx >> 4;
      const int c4  = (idx & 15) * 4;
      *(float4*)&k_lds[row][c4] =
          *(const float4*)&Km[base + (size_t)(jt * 16 + row) * HDIM + c4];
      *(float4*)&v_lds[row][c4] =
          *(const float4*)&Vm[base + (size_t)(jt * 16 + row) * HDIM + c4];
    }
    __syncthreads();

    // S = Q * K^T  (16 wmma steps over head_dim)
    v8f s = {0,0,0,0,0,0,0,0};
#pragma unroll
    for (int kk = 0; kk < 16; ++kk) {
      v2f bf;   // B[k][n] = K[n][k], n = Lm
      bf.x = k_lds[Lm][kk * 4 + 2 * half + 0];
      bf.y = k_lds[Lm][kk * 4 + 2 * half + 1];
      s = wmma_f32(qf[kk], bf, s);
    }

    const bool diag = (jt == qt);
#pragma unroll
    for (int r = 0; r < 8; ++r) {
      float sv = s[r] * 0.125f;  // 1/sqrt(64)
      if (diag && (jt * 16 + Lm > qt * 16 + r + 8 * half)) sv = -INFINITY;
      // row max across the 16-lane half-group (row lives in one half-wave)
      float rmax = sv;
      rmax = fmaxf(rmax, __shfl_xor(rmax, 1));
      rmax = fmaxf(rmax, __shfl_xor(rmax, 2));
      rmax = fmaxf(rmax, __shfl_xor(rmax, 4));
      rmax = fmaxf(rmax, __shfl_xor(rmax, 8));
      const float mnew = fmaxf(mrow[r], rmax);
      const float p = __expf(sv - mnew);
      float rsum = p;
      rsum += __shfl_xor(rsum, 1);
      rsum += __shfl_xor(rsum, 2);
      rsum += __shfl_xor(rsum, 4);
      rsum += __shfl_xor(rsum, 8);
      const float corr = __expf(mrow[r] - mnew);
      lrow[r] = lrow[r] * corr + rsum;
      mrow[r] = mnew;
      acc[0][r] *= corr; acc[1][r] *= corr;
      acc[2][r] *= corr; acc[3][r] *= corr;
      p_lds[r + 8 * half][Lm] = p;   // C-layout -> LDS
    }
    __syncthreads();

    // acc += P * V   (4 d-tiles x 4 wmma steps over 16 keys)
#pragma unroll
    for (int dt = 0; dt < 4; ++dt) {
#pragma unroll
      for (int kk = 0; kk < 4; ++kk) {
        v2f a;   // A-layout read of P
        a.x = p_lds[Lm][kk * 4 + 2 * half + 0];
        a.y = p_lds[Lm][kk * 4 + 2 * half + 1];
        v2f bf;  // B[k][d] = V[key k][d]
        bf.x = v_lds[kk * 4 + 2 * half + 0][dt * 16 + Lm];
        bf.y = v_lds[kk * 4 + 2 * half + 1][dt * 16 + Lm];
        acc[dt] = wmma_f32(a, bf, acc[dt]);
      }
    }
    __syncthreads();   // before next tile overwrites k_lds/v_lds/p_lds
  }

  // normalize and write ctx in [b, n, h*64+d] layout (ready for out-proj GEMM)
#pragma unroll
  for (int r = 0; r < 8; ++r) {
    const int qrow = qt * 16 + r + 8 * half;
    const float inv = 1.0f / lrow[r];
#pragma unroll
    for (int dt = 0; dt < 4; ++dt) {
      ctx[(size_t)(b * SEQ + qrow) * D_MODEL + h * HDIM + dt * 16 + Lm] =
          acc[dt][r] * inv;
    }
  }
}

// ---------------------------------------------------------------------------
extern "C" void kernel_launch(void* const* d_in, const int* in_sizes, int n_in,
                              void* d_out, int out_size, void* d_ws, size_t ws_size,
                              hipStream_t stream) {
  const float* x  = (const float*)d_in[0];
  const float* wq = (const float*)d_in[1];
  const float* wk = (const float*)d_in[2];
  const float* wv = (const float*)d_in[3];
  const float* wo = (const float*)d_in[4];
  float* out = (float*)d_out;

  const size_t mat = (size_t)ROWS * D_MODEL;   // 8192*768 floats
  float* q   = (float*)d_ws;
  float* k   = q + mat;
  float* v   = k + mat;
  float* ctx = v + mat;

  dim3 gGemm(D_MODEL / GEMM_BN, ROWS / 64);   // 6 x 128
  dim3 bGemm(256);
  // QKV projections, written directly in [b,h,n,d] layout
  gemm768_wmma<<<gGemm, bGemm, 0, stream>>>(x, wq, q, 1);
  gemm768_wmma<<<gGemm, bGemm, 0, stream>>>(x, wk, k, 1);
  gemm768_wmma<<<gGemm, bGemm, 0, stream>>>(x, wv, v, 1);

  dim3 gAttn(SEQ / 16, BATCH * HEADS);   // 128 x 48
  flash_attn_wmma<<<gAttn, dim3(32), 0, stream>>>(q, k, v, ctx);

  // output projection
  gemm768_wmma<<<gGemm, bGemm, 0, stream>>>(ctx, wo, out, 0);
}